// Demosaicing_71107478552657
// MI455X (gfx1250) — compile-verified
//
#include <hip/hip_runtime.h>
#include <stddef.h>
#include <stdint.h>

// ---------------------------------------------------------------------------
// Bilinear RGGB demosaic, MI455X (gfx1250).
// Memory-bound: 128 MiB in + 384 MiB out => ~22 us roofline @ 23.3 TB/s.
// CDNA5 path: async global->LDS DMA staging (ASYNCcnt) + LDS stencil compute.
// ---------------------------------------------------------------------------

#define IMG_H   2048
#define IMG_W   2048
#define TILE_W  64
#define TILE_H  16
#define HALO_W  (TILE_W + 2)          // 66
#define HALO_H  (TILE_H + 2)          // 18
#define LSTR    68                    // padded LDS row stride (floats)
#define NLOAD   (HALO_W * HALO_H)     // 1188 staged elements per block

#if defined(__gfx1250__) && __has_builtin(__builtin_amdgcn_global_load_async_to_lds_b32)
#define HAVE_ASYNC_LDS 1
#else
#define HAVE_ASYNC_LDS 0
#endif

typedef __attribute__((address_space(1))) int GInt;   // global (AS1) int
typedef __attribute__((address_space(3))) int LInt;   // LDS    (AS3) int

__global__ __launch_bounds__(256)
void demosaic_rggb_kernel(const float* __restrict__ x, float* __restrict__ out) {
    __shared__ float smem[HALO_H * LSTR];

    const int tid = (int)threadIdx.x;
    const int gx0 = (int)blockIdx.x * TILE_W;
    const int gy0 = (int)blockIdx.y * TILE_H;
    const int n   = (int)blockIdx.z;

    const float* img = x + (size_t)n * IMG_H * IMG_W;

    // ---- Stage tile + 1px halo into LDS (async DMA where available) ----
#pragma unroll
    for (int i = 0; i < (NLOAD + 255) / 256; ++i) {
        const int idx = tid + i * 256;
        if (idx < NLOAD) {
            const int r  = idx / HALO_W;
            const int c  = idx - r * HALO_W;
            const int gy = gy0 - 1 + r;
            const int gx = gx0 - 1 + c;
            const int l  = r * LSTR + c;
            if ((unsigned)gy < (unsigned)IMG_H && (unsigned)gx < (unsigned)IMG_W) {
#if HAVE_ASYNC_LDS
                __builtin_amdgcn_global_load_async_to_lds_b32(
                    (GInt*)(img + (size_t)gy * IMG_W + gx),
                    (LInt*)(&smem[l]),
                    /*imm offset*/ 0, /*cpol*/ 0);
#else
                smem[l] = img[(size_t)gy * IMG_W + gx];
#endif
            } else {
                smem[l] = 0.0f;   // zero padding == reference "SAME" conv border
            }
        }
    }
#if HAVE_ASYNC_LDS
#if __has_builtin(__builtin_amdgcn_s_wait_asynccnt)
    __builtin_amdgcn_s_wait_asynccnt(0);
#else
    asm volatile("s_wait_asynccnt 0" ::: "memory");
#endif
#endif
    __syncthreads();

    // ---- Each thread computes one 2x2 RGGB cell (4 pixels x 3 channels) ----
    const int tx = tid & 31;   // 0..31 -> 64 cols
    const int ty = tid >> 5;   // 0..7  -> 16 rows
    const int ly = 2 * ty + 1; // LDS row of cell origin (R site)
    const int lx = 2 * tx + 1;

    // v[r][c] = input at relative offset (r-1, c-1) from the R site
    float v[4][4];
#pragma unroll
    for (int r = 0; r < 4; ++r)
#pragma unroll
        for (int c = 0; c < 4; ++c)
            v[r][c] = smem[(ly - 1 + r) * LSTR + (lx - 1 + c)];

    // RGGB phases; weights from KRB=[[1,2,1],[2,4,2],[1,2,1]]/4, KG=cross/4,
    // applied to the parity-masked color planes (exact reference math).
    // Pixel A = R site (0,0)
    const float R_A = v[1][1];
    const float G_A = 0.25f * (v[1][0] + v[1][2] + v[0][1] + v[2][1]);
    const float B_A = 0.25f * (v[0][0] + v[0][2] + v[2][0] + v[2][2]);
    // Pixel B = Gr site (0,1)
    const float R_B = 0.5f * (v[1][1] + v[1][3]);
    const float G_B = v[1][2];
    const float B_B = 0.5f * (v[0][2] + v[2][2]);
    // Pixel C = Gb site (1,0)
    const float R_C = 0.5f * (v[1][1] + v[3][1]);
    const float G_C = v[2][1];
    const float B_C = 0.5f * (v[2][0] + v[2][2]);
    // Pixel D = B site (1,1)
    const float R_D = 0.25f * (v[1][1] + v[1][3] + v[3][1] + v[3][3]);
    const float G_D = 0.25f * (v[1][2] + v[2][1] + v[2][3] + v[3][2]);
    const float B_D = v[2][2];

    // ---- Vectorized float2 stores (x even -> 8B aligned, coalesced) ----
    const int gy = gy0 + 2 * ty;
    const int gx = gx0 + 2 * tx;
    const size_t plane = (size_t)IMG_H * IMG_W;
    float* o = out + (size_t)n * 3 * plane;
    const size_t row0 = (size_t)gy * IMG_W + gx;
    const size_t row1 = row0 + IMG_W;

    *(float2*)(o + 0 * plane + row0) = make_float2(R_A, R_B);
    *(float2*)(o + 0 * plane + row1) = make_float2(R_C, R_D);
    *(float2*)(o + 1 * plane + row0) = make_float2(G_A, G_B);
    *(float2*)(o + 1 * plane + row1) = make_float2(G_C, G_D);
    *(float2*)(o + 2 * plane + row0) = make_float2(B_A, B_B);
    *(float2*)(o + 2 * plane + row1) = make_float2(B_C, B_D);
}

extern "C" void kernel_launch(void* const* d_in, const int* in_sizes, int n_in,
                              void* d_out, int out_size, void* d_ws, size_t ws_size,
                              hipStream_t stream) {
    (void)n_in; (void)out_size; (void)d_ws; (void)ws_size;
    const float* x = (const float*)d_in[0];
    float* out = (float*)d_out;
    const int nImages = in_sizes[0] / (IMG_H * IMG_W);   // 8
    dim3 grid(IMG_W / TILE_W, IMG_H / TILE_H, nImages);  // 32 x 128 x 8
    demosaic_rggb_kernel<<<grid, 256, 0, stream>>>(x, out);
}